// RoIPool_4853313044500
// MI455X (gfx1250) — compile-verified
//
#include <hip/hip_runtime.h>
#include <hip/hip_bf16.h>
#include <stdint.h>

// RoI max pooling, tuned for MI455X (gfx1250, wave32).
// features: [N=4, C=512, H=64, W=64] f32   (32 MB -> L2 resident)
// rois:     [R, 5] f32 (b, x1, y1, x2, y2)
// out:      [R, 512, 7, 7] f32
//
// One block handles (one ROI) x (32 channels). Threads iterate bin-fastest so
// output stores are coalesced and a wave's loads span at most two 16KB channel
// planes. ROI row is staged into LDS via the CDNA5 async global->LDS path
// (ASYNCcnt), bin tables are computed once per block into LDS.

#define P_POOL 7
#define CHN    512
#define FH     64
#define FW     64
#define RSCALE (1.0f / 16.0f)

constexpr int CB    = 32;    // channels per block
constexpr int BLOCK = 256;   // 8 waves (wave32)

__global__ __launch_bounds__(BLOCK) void roipool_kernel(
    const float* __restrict__ features,
    const float* __restrict__ rois,
    float* __restrict__ out,
    int R)
{
    const int blocks_per_roi = CHN / CB;                 // 16
    const int r  = (int)(blockIdx.x / blocks_per_roi);
    const int c0 = (int)(blockIdx.x % blocks_per_roi) * CB;
    if (r >= R) return;

    __shared__ float s_roi[8];
    __shared__ int   s_hs[P_POOL], s_he[P_POOL], s_ws[P_POOL], s_we[P_POOL];

    const unsigned tid = threadIdx.x;

    // --- Stage the 5 ROI floats into LDS with gfx1250 async load-to-LDS. ---
    if (tid < 5u) {
        const float* gp = rois + (size_t)r * 5 + tid;
        // generic pointer to __shared__: low 32 bits are the wave-relative LDS
        // byte offset (LDS aperture keeps offset in addr[31:0]).
        unsigned lds_off = (unsigned)(uintptr_t)(&s_roi[tid]);
        asm volatile("global_load_async_to_lds_b32 %0, %1, off"
                     :
                     : "v"(lds_off), "v"(gp)
                     : "memory");
    }
    asm volatile("s_wait_asynccnt 0" ::: "memory");
    __syncthreads();

    // --- Block-uniform ROI parameters (match JAX f32 semantics exactly). ---
    const int b  = (int)s_roi[0];
    const int xs = (int)rintf(s_roi[1] * RSCALE);   // RNE == jnp.round
    const int ys = (int)rintf(s_roi[2] * RSCALE);
    const int xe = (int)rintf(s_roi[3] * RSCALE);
    const int ye = (int)rintf(s_roi[4] * RSCALE);
    const float roi_w = fmaxf((float)(xe - xs + 1), 1.0f);
    const float roi_h = fmaxf((float)(ye - ys + 1), 1.0f);
    const float bin_w = roi_w / 7.0f;
    const float bin_h = roi_h / 7.0f;

    if (tid < (unsigned)P_POOL) {
        const float p = (float)tid;
        int hs = (int)floorf(p * bin_h) + ys;
        int he = (int)ceilf((p + 1.0f) * bin_h) + ys;
        int ws = (int)floorf(p * bin_w) + xs;
        int we = (int)ceilf((p + 1.0f) * bin_w) + xs;
        s_hs[tid] = min(max(hs, 0), FH);
        s_he[tid] = min(max(he, 0), FH);
        s_ws[tid] = min(max(ws, 0), FW);
        s_we[tid] = min(max(we, 0), FW);
    }
    __syncthreads();

    const float* fbase = features + ((size_t)b * CHN + (size_t)c0) * (FH * FW);
    float*       obase = out      + ((size_t)r * CHN + (size_t)c0) * (P_POOL * P_POOL);

    // bin index fastest -> coalesced stores; all lanes of a wave read within
    // at most two 16KB channel planes (L2-resident).
    for (int idx = (int)tid; idx < CB * P_POOL * P_POOL; idx += BLOCK) {
        const int c   = idx / (P_POOL * P_POOL);
        const int bin = idx % (P_POOL * P_POOL);
        const int ph  = bin / P_POOL;
        const int pw  = bin % P_POOL;

        const int hs = s_hs[ph], he = s_he[ph];
        const int ws = s_ws[pw], we = s_we[pw];

        float m;
        if (he <= hs || we <= ws) {
            m = 0.0f;                              // empty bin -> exact 0
        } else {
            m = -3.4e38f;
            const float* plane = fbase + (size_t)c * (FH * FW);
            for (int h = hs; h < he; ++h) {
                const float* row = plane + h * FW;
                for (int w = ws; w < we; ++w) {
                    m = fmaxf(m, row[w]);
                }
            }
        }
        obase[idx] = m;
    }
}

extern "C" void kernel_launch(void* const* d_in, const int* in_sizes, int n_in,
                              void* d_out, int out_size, void* d_ws, size_t ws_size,
                              hipStream_t stream) {
    const float* features = (const float*)d_in[0];
    const float* rois     = (const float*)d_in[1];
    float*       out      = (float*)d_out;

    const int R = in_sizes[1] / 5;                     // rois is [R,5]
    const int blocks = R * (CHN / CB);                 // one block per (roi, 32-ch tile)

    roipool_kernel<<<blocks, BLOCK, 0, stream>>>(features, rois, out, R);
}